// custom_v5_180388626854
// MI455X (gfx1250) — compile-verified
//
#include <hip/hip_runtime.h>

typedef __attribute__((ext_vector_type(2))) float v2f;
typedef __attribute__((ext_vector_type(8))) float v8f;

#define B_   32
#define T_   32
#define R_   256
#define YS_  64
#define OT   32             // output tile edge
#define HT_  34             // hidden tile edge (OT+2)
#define XT_  36             // x tile edge (OT+4)
#define HPIX (HT_*HT_)      // 1156 hidden pixels
#define XPIX (XT_*XT_)      // 1296 x pixels
#define NT1  ((HPIX+15)/16) // 73 pixel tiles for conv1
#define HP2  1168           // padded float2 stride per channel-pair (mod 32 == 16)
#define XPAIR (XPIX/2)      // 648 even-q pixel pairs in the x tile

__global__ __launch_bounds__(256)
void fused_recon_kernel(const float* __restrict__ yt,
                        const float* __restrict__ Ht,
                        const float* __restrict__ W1,
                        const float* __restrict__ b1,
                        const float* __restrict__ W2,
                        const float* __restrict__ b2,
                        float* __restrict__ out)
{
    __shared__ float  xs[XPIX];        // einsum tile (halo 2)
    __shared__ float2 hs2[16 * HP2];   // hidden, pair-channel layout (~146KB)
    __shared__ float  w1s[32 * 12];    // W1, K padded 9->12
    __shared__ float  w2p[288];        // W2 permuted tap-major: w2p[tap*32+c]
    __shared__ float  b1s[32];

    const int tid = threadIdx.x;
    const int bat = blockIdx.y;
    const int p0  = (blockIdx.x >> 3) * OT;
    const int q0  = (blockIdx.x & 7) * OT;

    // ---- weights / biases into LDS ----
    if (tid < 32 * 12) {
        int c = tid / 12, k = tid % 12;
        w1s[tid] = (k < 9) ? W1[c * 9 + k] : 0.0f;
    }
    for (int j = tid; j < 288; j += 256) {      // tap-major permute of W2
        int tap = j >> 5, c = j & 31;
        w2p[j] = W2[c * 9 + tap];
    }
    if (tid < 32) b1s[tid] = b1[tid];
    const float b2v = b2[0];

    // ---- stage 0: x[p,q] = sum_t Ht[t,p,q] * yt[b,t,p/4,q/4], 36x36 tile ----
    // Even-q pixel pairs: 8B-aligned b64 Ht loads, one shared yt load per pair.
    const float* ytb = yt + (size_t)bat * T_ * YS_ * YS_;
    for (int i = tid; i < XPAIR; i += 256) {
        const int lp  = i / (XT_ / 2);
        const int lq2 = (i - lp * (XT_ / 2)) * 2;
        const int p = p0 + lp - 2;
        const int q = q0 + lq2 - 2;             // even; pair q,q+1 shares q>>2
        float v0 = 0.0f, v1 = 0.0f;
        if (p >= 0 && p < R_ && q >= 0) {       // q+1 can reach 257 at the right tile edge
            const float2* hp = (const float2*)(Ht + p * R_ + q);
            const float*  yp = ytb + (p >> 2) * YS_ + (q >> 2);
#pragma unroll
            for (int t = 0; t < T_; ++t) {
                const float2 hv = hp[t * (R_ * R_ / 2)];
                const float  yv = yp[t * YS_ * YS_];
                v0 = fmaf(hv.x, yv, v0);
                v1 = fmaf(hv.y, yv, v1);
            }
        }
        *(float2*)&xs[lp * XT_ + lq2] = make_float2(v0, v1);
    }
    __syncthreads();

    const int wave   = tid >> 5;
    const int lane   = tid & 31;
    const int lane15 = lane & 15;
    const int khi    = (lane >= 16) ? 2 : 0;  // A/B frag: hi half-wave holds K=2,3
    const int hi8    = (lane >= 16) ? 8 : 0;  // C/D frag: hi half-wave holds M=8..15

    // ---- stage 1: conv1 (1->32, 3x3) + bias + ReLU via v_wmma_f32_16x16x4_f32 ----
    for (int tile = wave; tile < 2 * NT1; tile += 8) {
        const int mt = tile & 1;               // channel tile
        const int nt = tile >> 1;              // pixel tile
        const int m  = mt * 16 + lane15;       // A row = channel
        const int n  = nt * 16 + lane15;       // B col = hidden pixel
        const int nn = (n < HPIX) ? n : (HPIX - 1);
        const int hp = nn / HT_, hq = nn % HT_;

        v8f acc;
#pragma unroll
        for (int r = 0; r < 8; ++r) acc[r] = b1s[mt * 16 + hi8 + r];

#pragma unroll
        for (int kc = 0; kc < 3; ++kc) {
            const int k0 = kc * 4 + khi;
            const int k1 = k0 + 1;
            v2f a, bf;
            a[0] = w1s[m * 12 + k0];
            a[1] = w1s[m * 12 + k1];
            const int t0 = (k0 < 9) ? k0 : 8;  // padded taps: a==0, value irrelevant
            const int t1 = (k1 < 9) ? k1 : 8;
            bf[0] = xs[(hp + t0 / 3) * XT_ + hq + t0 % 3];
            bf[1] = xs[(hp + t1 / 3) * XT_ + hq + t1 % 3];
            acc = __builtin_amdgcn_wmma_f32_16x16x4_f32(false, a, false, bf,
                                                        (short)0, acc, false, false);
        }
        if (n < HPIX) {
            const int cb2 = (mt * 16 + hi8) >> 1;      // channel-pair base
#pragma unroll
            for (int rp = 0; rp < 4; ++rp) {           // paired-channel b64 stores
                const float v0 = fmaxf(acc[2 * rp], 0.0f);       // ReLU -> v_max_num
                const float v1 = fmaxf(acc[2 * rp + 1], 0.0f);
                hs2[(cb2 + rp) * HP2 + n] = make_float2(v0, v1);
            }
        }
    }
    __syncthreads();

    // ---- stage 2: conv2 (32->1, 3x3) via WMMA; K = tap*32 + c (tap-major) ----
    // A = h patches (16 pixels x 288), B = W2 replicated across 16 columns.
    float* outb = out + (size_t)bat * R_ * R_;
    for (int g = wave; g < (OT * OT) / 16; g += 8) {
        const int pix = g * 16 + lane15;       // A row = output pixel in tile
        const int op = pix >> 5, oq = pix & 31;
        v8f acc0, acc1;
#pragma unroll
        for (int r = 0; r < 8; ++r) { acc0[r] = b2v; acc1[r] = 0.0f; }

        for (int tap = 0; tap < 9; ++tap) {
            const int dy = tap / 3, dx = tap % 3;           // tap-loop invariant
            const int npix = (op + dy) * HT_ + (oq + dx);   // pixel for this tap
            const float2* __restrict__ hrow = hs2 + npix;
            const float2* __restrict__ wrow = (const float2*)(w2p + tap * 32);
#pragma unroll
            for (int cc = 0; cc < 8; ++cc) {
                const int cp = cc * 2 + (khi >> 1);   // channel-pair index
                const float2 av = hrow[cp * HP2];     // one ds_load_b64
                const float2 bv = wrow[cp];           // one ds_load_b64 (broadcast)
                v2f a, bf;
                a[0] = av.x; a[1] = av.y;
                bf[0] = bv.x; bf[1] = bv.y;
                if ((cc & 1) == 0)
                    acc0 = __builtin_amdgcn_wmma_f32_16x16x4_f32(false, a, false, bf,
                                                                 (short)0, acc0, false, false);
                else
                    acc1 = __builtin_amdgcn_wmma_f32_16x16x4_f32(false, a, false, bf,
                                                                 (short)0, acc1, false, false);
            }
        }
        if (lane15 == 0) {   // column N=0 of D carries the result; 8 consecutive pixels
            const int pix2 = g * 16 + hi8;
            const int op2 = pix2 >> 5, oq2 = pix2 & 31;
            float s[8];
#pragma unroll
            for (int r = 0; r < 8; ++r) s[r] = acc0[r] + acc1[r];
            float4* dst = (float4*)&outb[(p0 + op2) * R_ + (q0 + oq2)];
            dst[0] = make_float4(s[0], s[1], s[2], s[3]);
            dst[1] = make_float4(s[4], s[5], s[6], s[7]);
        }
    }
}

extern "C" void kernel_launch(void* const* d_in, const int* in_sizes, int n_in,
                              void* d_out, int out_size, void* d_ws, size_t ws_size,
                              hipStream_t stream) {
    (void)in_sizes; (void)n_in; (void)d_ws; (void)ws_size; (void)out_size;
    const float* yt = (const float*)d_in[0];
    const float* Ht = (const float*)d_in[1];
    const float* W1 = (const float*)d_in[2];
    const float* b1 = (const float*)d_in[3];
    const float* W2 = (const float*)d_in[4];
    const float* b2 = (const float*)d_in[5];
    float* out = (float*)d_out;

    dim3 grid(64, 32);   // 8x8 tiles of 32x32, 32 batches
    fused_recon_kernel<<<grid, 256, 0, stream>>>(yt, Ht, W1, b1, W2, b2, out);
}